// Renderer_30880814859163
// MI455X (gfx1250) — compile-verified
//
#include <hip/hip_runtime.h>
#include <math.h>

// CDNA5 (gfx1250) depth rasterizer, v4.
// Edge functions + depth plane are affine in pixel coords -> evaluating
// {b0,b1,b2,zinv} for 16 pixels x 4 triangles is a K=3 GEMM:
// V_WMMA_F32_16X16X4_F32 (f32 matrix pipe, wave32).
// v4: conservative per-group screen bboxes + wave-uniform (SGPR-branch) tile
// culling: skipped groups cost ~8 instructions instead of 16 WMMAs + ~400 VALU.
// EXEC stays all-ones across the cull branch (readfirstlane -> s_cbranch),
// as required by WMMA.

typedef __attribute__((ext_vector_type(2))) float v2f;
typedef __attribute__((ext_vector_type(8))) float v8f;

#define IMG    256       // supersampled resolution (2x AA)
#define OUTN   128
#define NTRI   4096      // 2048 faces x 2 windings
#define NGROUP (NTRI / 4)
#define NCHUNK 32
#define GPC    (NGROUP / NCHUNK)   // 32 groups (128 triangles) per chunk
#define NEARP  0.1f
#define FARP   100.0f
#define ZINV_LO 0.01f    // 1/FAR   (z < FAR  <=> zinv > 1/FAR)
#define ZINV_HI 10.0f    // 1/NEAR  (z > NEAR <=> zinv < 1/NEAR)
#define FARBITS 0x42C80000u        // __float_as_uint(100.0f)

// ---------------- Kernel 1: vertex projection ----------------
__global__ __launch_bounds__(256) void project_kernel(
    const float* __restrict__ verts,  // 4096*3
    const float* __restrict__ Km,     // 3x3
    const float* __restrict__ Rm,     // 3x3
    const float* __restrict__ tm,     // 3
    const float* __restrict__ dc,     // 5
    float4* __restrict__ proj)        // 4096 x (u, v, z, 0)
{
    int i = blockIdx.x * blockDim.x + threadIdx.x;
    if (i >= 4096) return;
    float vx = verts[3*i+0], vy = verts[3*i+1], vz = verts[3*i+2];
    float x = Rm[0]*vx + Rm[1]*vy + Rm[2]*vz + tm[0];
    float y = Rm[3]*vx + Rm[4]*vy + Rm[5]*vz + tm[1];
    float z = Rm[6]*vx + Rm[7]*vy + Rm[8]*vz + tm[2];
    float iz = 1.0f / (z + 1e-5f);
    float x_ = x * iz, y_ = y * iz;
    float r2 = x_*x_ + y_*y_;
    float k1 = dc[0], k2 = dc[1], p1 = dc[2], p2 = dc[3], k3 = dc[4];
    float radial = 1.0f + r2*(k1 + r2*(k2 + r2*k3));
    float xy = x_*y_;
    float xd = x_*radial + 2.0f*p1*xy + p2*(r2 + 2.0f*x_*x_);
    float yd = y_*radial + p1*(r2 + 2.0f*y_*y_) + 2.0f*p2*xy;
    float u  = Km[0]*xd + Km[1]*yd + Km[2];
    float vv = Km[3]*xd + Km[4]*yd + Km[5];
    vv = 1024.0f - vv;
    u  = (u  - 512.0f) * (2.0f / 1024.0f);
    vv = (vv - 512.0f) * (2.0f / 1024.0f);
    proj[i] = make_float4(u, vv, z, 0.0f);
}

// ---------------- Kernel 2: per-triangle plane setup + bbox ----------------
// 4 rows per triangle: (const, x-coef, y-coef, 0) for b0,b1,b2,zinv,
// pre-scaled by 1/area. Degenerate triangles get b0 == -1 and an empty bbox.
// bbox is in pixel-index space: pixel ix has u = (2*ix+1-256)/256, i.e.
// ix(u) = 128*u + 127.5 ; pixel iy has v = (255-2*iy)/256, i.e.
// iy(v) = 127.5 - 128*v (decreasing in v).
__global__ __launch_bounds__(256) void setup_kernel(
    const int* __restrict__ faces,    // 2048*3
    const float4* __restrict__ proj,
    float4* __restrict__ rows,        // NTRI*4
    float4* __restrict__ tribbox)     // NTRI x (ixmin, ixmax, iymin, iymax)
{
    int t = blockIdx.x * blockDim.x + threadIdx.x;
    if (t >= NTRI) return;
    int f = t & 2047;
    int rev = (t >= 2048);
    int i0 = faces[3*f + (rev ? 2 : 0)];
    int i1 = faces[3*f + 1];
    int i2 = faces[3*f + (rev ? 0 : 2)];
    float4 P0 = proj[i0], P1 = proj[i1], P2 = proj[i2];

    float A0 = P1.x*P2.y - P2.x*P1.y, B0 = P1.y - P2.y, C0 = P2.x - P1.x;
    float A1 = P2.x*P0.y - P0.x*P2.y, B1 = P2.y - P0.y, C1 = P0.x - P2.x;
    float A2 = P0.x*P1.y - P1.x*P0.y, B2 = P0.y - P1.y, C2 = P1.x - P0.x;
    float area = A0 + A1 + A2;            // x/y coefs cancel: constant
    bool ok = fabsf(area) > 1e-10f;
    float inv = ok ? 1.0f / area : 1.0f;
    float q0 = inv / P0.z, q1 = inv / P1.z, q2 = inv / P2.z;

    float4 r0 = ok ? make_float4(A0*inv, B0*inv, C0*inv, 0.0f)
                   : make_float4(-1.0f, 0.0f, 0.0f, 0.0f);  // never inside
    float4 r1 = make_float4(A1*inv, B1*inv, C1*inv, 0.0f);
    float4 r2 = make_float4(A2*inv, B2*inv, C2*inv, 0.0f);
    float4 r3 = make_float4(A0*q0 + A1*q1 + A2*q2,
                            B0*q0 + B1*q1 + B2*q2,
                            C0*q0 + C1*q1 + C2*q2, 0.0f);
    rows[4*t+0] = r0; rows[4*t+1] = r1; rows[4*t+2] = r2; rows[4*t+3] = r3;

    float umin = fminf(P0.x, fminf(P1.x, P2.x));
    float umax = fmaxf(P0.x, fmaxf(P1.x, P2.x));
    float vmin = fminf(P0.y, fminf(P1.y, P2.y));
    float vmax = fmaxf(P0.y, fmaxf(P1.y, P2.y));
    float4 bb;
    bb.x = ok ? (128.0f * umin + 127.5f) : 1e30f;    // ixmin
    bb.y = ok ? (128.0f * umax + 127.5f) : -1e30f;   // ixmax
    bb.z = ok ? (127.5f - 128.0f * vmax) : 1e30f;    // iymin (v decreasing)
    bb.w = ok ? (127.5f - 128.0f * vmin) : -1e30f;   // iymax
    tribbox[t] = bb;
}

// ---------------- Kernel 3: merge 4 triangle bboxes -> 1 group bbox --------
__global__ __launch_bounds__(256) void groupbbox_kernel(
    const float4* __restrict__ tribbox,
    float4* __restrict__ gbbox)       // NGROUP
{
    int g = blockIdx.x * blockDim.x + threadIdx.x;
    if (g >= NGROUP) return;
    float4 a = tribbox[4*g+0], b = tribbox[4*g+1];
    float4 c = tribbox[4*g+2], d = tribbox[4*g+3];
    float4 r;
    r.x = fminf(fminf(a.x, b.x), fminf(c.x, d.x));
    r.y = fmaxf(fmaxf(a.y, b.y), fmaxf(c.y, d.y));
    r.z = fminf(fminf(a.z, b.z), fminf(c.z, d.z));
    r.w = fmaxf(fmaxf(a.w, b.w), fmaxf(c.w, d.w));
    gbbox[g] = r;
}

// ---------------- Kernel 4: depth buffer init ----------------
__global__ __launch_bounds__(256) void init_depth_kernel(unsigned int* __restrict__ depthbuf)
{
    int i = blockIdx.x * blockDim.x + threadIdx.x;
    if (i < IMG * IMG) depthbuf[i] = FARBITS;
}

// ---------------- Kernel 5: WMMA rasterizer ----------------
// One wave32 per (16x16 pixel tile, 128-triangle chunk). Block = 8 waves
// sharing one chunk: 8 KB of plane rows staged in LDS once.
// grid.x = 32 tile-blocks * 32 chunks = 1024 blocks -> 8192 waves.
__global__ __launch_bounds__(256) void raster_kernel(
    const float4* __restrict__ rows,
    const float4* __restrict__ gbbox,
    unsigned int* __restrict__ depthbuf)   // 256x256 pre-AA, u32 depth bits
{
    __shared__ float4 srows[GPC * 16];     // 512 rows = 8 KB

    const int tid   = threadIdx.x;
    const int chunk = blockIdx.x >> 5;     // 0..31
    const int tb    = blockIdx.x & 31;     // tile-block 0..31

    // Cooperative stage: 512 float4 rows, 2 per thread.
    srows[tid]       = rows[chunk * (GPC * 16) + tid];
    srows[tid + 256] = rows[chunk * (GPC * 16) + tid + 256];
    __syncthreads();

    const int lane = tid & 31;
    const int wave = tid >> 5;
    const int tile = tb * 8 + wave;        // 0..255
    const int tx = (tile & 15) * 16;
    const int ty = (tile >> 4) * 16;
    const int half = lane >> 4;
    const int l    = lane & 15;

    // Pixel basis B-fragments (ISA 32-bit 4x16 B layout: reg0=K0|K2, reg1=K1|K3).
    const float xc = (2.0f * (float)(tx + l) + 1.0f - 256.0f) * (1.0f / 256.0f);
    v2f bfrag[16];
#pragma unroll
    for (int s = 0; s < 16; ++s) {
        float yc = (255.0f - 2.0f * (float)(ty + s)) * (1.0f / 256.0f);
        if (half) { bfrag[s].x = yc;   bfrag[s].y = 0.0f; }   // (K2, K3)
        else      { bfrag[s].x = 1.0f; bfrag[s].y = xc;   }   // (K0, K1)
    }

    // Max of valid zinv per pixel row (0 = "nothing hit": all valid zinv > 1/FAR).
    float zmax[16];
#pragma unroll
    for (int s = 0; s < 16; ++s) zmax[s] = 0.0f;

    const char* sbase = (const char*)srows;
    const v8f czero = {};
    const float txlo = (float)tx,        txhi = (float)(tx + 15);
    const float tylo = (float)ty,        tyhi = (float)(ty + 15);

    for (int g = 0; g < GPC; ++g) {
        // Wave-uniform conservative bbox cull (SGPR branch; EXEC untouched).
        float4 bb = gbbox[chunk * GPC + g];
        bool ov = (bb.x <= txhi) && (bb.y >= txlo) &&
                  (bb.z <= tyhi) && (bb.w >= tylo);
        if (!__builtin_amdgcn_readfirstlane((int)ov)) continue;

        // A-fragment from LDS: 16 rows (4 tris x {b0,b1,b2,zinv}), K-striped:
        // lanes 0-15 read (c0,c1) of row l; lanes 16-31 read (c2, 0).
        v2f af = *(const v2f*)(sbase + (size_t)(g * 16 + l) * 16 + half * 8);

#pragma unroll
        for (int s = 0; s < 16; ++s) {
            v8f d = __builtin_amdgcn_wmma_f32_16x16x4_f32(
                false, af, false, bfrag[s],
                (short)0, czero, false, false);
            float zm = zmax[s];
#pragma unroll
            for (int tr = 0; tr < 2; ++tr) {
                float b0 = d[4*tr+0], b1 = d[4*tr+1];
                float b2 = d[4*tr+2], zi = d[4*tr+3];
                bool inside = (b0 >= 0.0f) && (b1 >= 0.0f) && (b2 >= 0.0f);
                float ze = (fabsf(zi) > 1e-10f) ? zi : 1.0f;  // |zinv| tiny -> z=1
                // NEAR < z < FAR  <=>  ZINV_LO < ze < ZINV_HI   (ze > 0 implied)
                bool valid = inside && (ze > ZINV_LO) && (ze < ZINV_HI);
                zm = fmaxf(zm, valid ? ze : 0.0f);
            }
            zmax[s] = zm;
        }
    }

    // Merge lane-halves; one exact divide per pixel; publish with u32 atomic-min
    // (bitwise order == float order for positive values -> exact + deterministic).
#pragma unroll
    for (int s = 0; s < 16; ++s) {
        float zm = fmaxf(zmax[s], __shfl_xor(zmax[s], 16, 32));
        if (half == 0 && zm > 0.0f) {
            float z = 1.0f / zm;   // min depth = 1 / max zinv
            atomicMin(&depthbuf[(ty + s) * IMG + (tx + l)], __float_as_uint(z));
        }
    }
}

// ---------------- Kernel 6: 2x2 AA downsample ----------------
__global__ __launch_bounds__(256) void downsample_kernel(
    const unsigned int* __restrict__ depthbuf,
    float* __restrict__ out)
{
    int idx = blockIdx.x * blockDim.x + threadIdx.x;
    if (idx >= OUTN * OUTN) return;
    int j = idx >> 7, i = idx & 127;
    const unsigned int* r0 = depthbuf + (2 * j) * IMG + 2 * i;
    const unsigned int* r1 = r0 + IMG;
    float a = __uint_as_float(r0[0]), b = __uint_as_float(r0[1]);
    float c = __uint_as_float(r1[0]), d = __uint_as_float(r1[1]);
    out[idx] = (a + b + c + d) * 0.25f;
}

extern "C" void kernel_launch(void* const* d_in, const int* in_sizes, int n_in,
                              void* d_out, int out_size, void* d_ws, size_t ws_size,
                              hipStream_t stream) {
    const float* vertices = (const float*)d_in[0];
    const int*   faces    = (const int*)d_in[1];
    const float* Km       = (const float*)d_in[2];
    const float* Rm       = (const float*)d_in[3];
    const float* tm       = (const float*)d_in[4];
    const float* dc       = (const float*)d_in[5];
    float* out = (float*)d_out;

    char* ws = (char*)d_ws;
    float4*       proj     = (float4*)ws;                               // 64 KB
    float4*       rows     = (float4*)(ws + 64 * 1024);                 // 256 KB
    unsigned int* depthbuf = (unsigned int*)(ws + (64 + 256) * 1024);   // 256 KB
    float4*       tribbox  = (float4*)(ws + (64 + 256 + 256) * 1024);   // 64 KB
    float4*       gbbox    = (float4*)(ws + (64 + 256 + 256 + 64) * 1024); // 16 KB

    project_kernel<<<16, 256, 0, stream>>>(vertices, Km, Rm, tm, dc, proj);
    setup_kernel<<<16, 256, 0, stream>>>(faces, proj, rows, tribbox);
    groupbbox_kernel<<<4, 256, 0, stream>>>(tribbox, gbbox);
    init_depth_kernel<<<(IMG * IMG) / 256, 256, 0, stream>>>(depthbuf);
    raster_kernel<<<32 * NCHUNK, 256, 0, stream>>>(rows, gbbox, depthbuf);
    downsample_kernel<<<(OUTN * OUTN) / 256, 256, 0, stream>>>(depthbuf, out);
}